// CRF_18983755448446
// MI455X (gfx1250) — compile-verified
//
#include <hip/hip_runtime.h>

namespace {
constexpr int kT = 1024;   // SEQ_LEN
constexpr int kB = 512;    // BATCH
constexpr int kC = 48;     // NC
}

typedef __attribute__((ext_vector_type(2))) float v2f;
typedef __attribute__((ext_vector_type(8))) float v8f;

// One wave (32 lanes) handles a 16-batch tile for the full 1024-step scan.
// logZ state is held in WMMA C-layout registers (3 tiles of v8f):
//   tile ct, vgpr r, lane l  ->  element (row = r + 8*(l>>4), col = 16*ct + (l&15))
__global__ __launch_bounds__(32)
void crf_fwd_kernel(const float* __restrict__ scores,
                    const int*   __restrict__ target,
                    const int*   __restrict__ lengths,
                    const float* __restrict__ transitions,
                    float*       __restrict__ out)
{
    __shared__ float aLds[16 * kC];   // staging for C-layout -> A-layout transpose
    __shared__ float rowLse[16];

    const int lane = threadIdx.x;     // 0..31
    const int n    = lane & 15;       // column within a 16-wide tile
    const int h    = lane >> 4;       // lane half
    const int base = blockIdx.x * 16; // first batch of this tile

    // ---- precompute exp(transitions) in WMMA B-layout registers ----
    // B chunk kc (K = 4kc..4kc+3), class tile ct:
    //   vgpr0: lanes 0-15 -> K=4kc+0, lanes 16-31 -> K=4kc+2
    //   vgpr1: lanes 0-15 -> K=4kc+1, lanes 16-31 -> K=4kc+3
    float bm0[12][3], bm1[12][3];
    #pragma unroll
    for (int kc = 0; kc < 12; ++kc) {
        #pragma unroll
        for (int ct = 0; ct < 3; ++ct) {
            const int k0 = 4 * kc + 2 * h;
            bm0[kc][ct] = __expf(transitions[(k0 + 0) * kC + 16 * ct + n]);
            bm1[kc][ct] = __expf(transitions[(k0 + 1) * kC + 16 * ct + n]);
        }
    }

    // per-row valid lengths (row = r + 8h)
    int lenr[8];
    #pragma unroll
    for (int r = 0; r < 8; ++r) lenr[r] = lengths[base + r + 8 * h];

    const int bl   = base + n;        // this lane's batch for the X (target) path
    const int llen = lengths[bl];

    // ---- t = 0 init: logZ = scores[0], X = scores[0][b, target[0,b]] ----
    v8f c[3];
    #pragma unroll
    for (int ct = 0; ct < 3; ++ct) {
        #pragma unroll
        for (int r = 0; r < 8; ++r)
            c[ct][r] = scores[(size_t)(base + r + 8 * h) * kC + 16 * ct + n];
    }

    int   prevT = target[bl];
    float X     = scores[(size_t)bl * kC + prevT];

    for (int t = 1; t < kT; ++t) {
        const float* sc_t = scores + (size_t)t * kB * kC;

        // Prefetch next timestep's slab for this tile (global_prefetch_b8).
        // Unconditional: speculative prefetch past the buffer end is dropped
        // silently by hardware, and this keeps the loop body un-duplicated.
        __builtin_prefetch(sc_t + (size_t)kB * kC + (size_t)base * kC + lane * 16, 0, 1);

        // scores for this step, directly in C-layout (read-once -> nontemporal)
        float sreg[3][8];
        #pragma unroll
        for (int ct = 0; ct < 3; ++ct) {
            #pragma unroll
            for (int r = 0; r < 8; ++r)
                sreg[ct][r] = __builtin_nontemporal_load(
                    sc_t + (size_t)(base + r + 8 * h) * kC + 16 * ct + n);
        }

        // target-score path (lanes 16-31 redundantly duplicate lanes 0-15)
        const int   tgt = target[(size_t)t * kB + bl];
        const float tv  = transitions[prevT * kC + tgt];
        const float ev  = sc_t[(size_t)bl * kC + tgt];
        if (t < llen) X += tv + ev;
        prevT = tgt;

        // ---- per-row max over 48 classes (tile-wise max, then xor-shuffle) ----
        float m[8];
        #pragma unroll
        for (int r = 0; r < 8; ++r) {
            float v = fmaxf(fmaxf(c[0][r], c[1][r]), c[2][r]);
            v = fmaxf(v, __shfl_xor(v, 1));
            v = fmaxf(v, __shfl_xor(v, 2));
            v = fmaxf(v, __shfl_xor(v, 4));
            v = fmaxf(v, __shfl_xor(v, 8));
            m[r] = v;
        }

        // ---- a = exp(logZ - m) staged to LDS (C-layout positions) ----
        #pragma unroll
        for (int ct = 0; ct < 3; ++ct) {
            #pragma unroll
            for (int r = 0; r < 8; ++r)
                aLds[(r + 8 * h) * kC + 16 * ct + n] = __expf(c[ct][r] - m[r]);
        }
        // single wave: LDS ops are in-order; compiler inserts s_wait_dscnt

        // ---- preload ALL 12 A-layout chunks into registers first, so the
        //      WMMA chain below runs back-to-back without dscnt stalls ----
        float a0[12], a1[12];
        #pragma unroll
        for (int kc = 0; kc < 12; ++kc) {
            a0[kc] = aLds[n * kC + 4 * kc + 2 * h + 0];
            a1[kc] = aLds[n * kC + 4 * kc + 2 * h + 1];
        }

        // ---- S = a(16x48) x expT(48x48) via 36x V_WMMA_F32_16X16X4_F32 ----
        v8f d[3] = {};
        #pragma unroll
        for (int kc = 0; kc < 12; ++kc) {
            v2f a;
            a.x = a0[kc];
            a.y = a1[kc];
            #pragma unroll
            for (int ct = 0; ct < 3; ++ct) {
                v2f b;
                b.x = bm0[kc][ct];
                b.y = bm1[kc][ct];
                d[ct] = __builtin_amdgcn_wmma_f32_16x16x4_f32(
                    false, a, false, b, (short)0, d[ct], false, false);
            }
        }

        // ---- logZ_new = s_t + m + log(S), frozen past each chain's length ----
        #pragma unroll
        for (int ct = 0; ct < 3; ++ct) {
            #pragma unroll
            for (int r = 0; r < 8; ++r) {
                const float nz = sreg[ct][r] + m[r] + __logf(d[ct][r]);
                c[ct][r] = (t < lenr[r]) ? nz : c[ct][r];
            }
        }
    }

    // ---- final logZ[b] = logsumexp over 48 classes ----
    float lse[8];
    #pragma unroll
    for (int r = 0; r < 8; ++r) {
        float v = fmaxf(fmaxf(c[0][r], c[1][r]), c[2][r]);
        v = fmaxf(v, __shfl_xor(v, 1));
        v = fmaxf(v, __shfl_xor(v, 2));
        v = fmaxf(v, __shfl_xor(v, 4));
        v = fmaxf(v, __shfl_xor(v, 8));
        const float M = v;
        float s = __expf(c[0][r] - M) + __expf(c[1][r] - M) + __expf(c[2][r] - M);
        s += __shfl_xor(s, 1);
        s += __shfl_xor(s, 2);
        s += __shfl_xor(s, 4);
        s += __shfl_xor(s, 8);
        lse[r] = M + __logf(s);
    }

    if (n == 0) {                       // lanes 0 and 16 publish their 8 rows
        #pragma unroll
        for (int r = 0; r < 8; ++r) rowLse[8 * h + r] = lse[r];
    }
    __syncthreads();                    // single-wave WG: effectively free

    if (lane < 16) {
        out[bl]      = X;               // target score
        out[kB + bl] = X - rowLse[n];   // log-likelihood
    }
}

extern "C" void kernel_launch(void* const* d_in, const int* in_sizes, int n_in,
                              void* d_out, int out_size, void* d_ws, size_t ws_size,
                              hipStream_t stream) {
    (void)in_sizes; (void)n_in; (void)out_size; (void)d_ws; (void)ws_size;
    const float* scores      = (const float*)d_in[0];
    const int*   target      = (const int*)d_in[1];
    const int*   lengths     = (const int*)d_in[2];
    const float* transitions = (const float*)d_in[3];
    float*       out         = (float*)d_out;

    crf_fwd_kernel<<<kB / 16, 32, 0, stream>>>(scores, target, lengths, transitions, out);
}